// ChiSq_773094113289
// MI455X (gfx1250) — compile-verified
//
#include <hip/hip_runtime.h>

// ---------------------------------------------------------------------------
// Power chi-square (PyCBC-style) for MI455X / gfx1250.
// One workgroup (256 threads = 8 wave32) per (batch, channel) row.
// Full pipeline fused in LDS: rFFT(16384 real) via 8192-pt complex FFT
// (16 x 512-pt radix-2 in LDS + radix-16 recombine as fp32 WMMA GEMM),
// untangle, cum_h scan, searchsorted edges, cum_q scan, chi-square.
// Rows are DMA'd global->LDS with GLOBAL_LOAD_ASYNC_TO_LDS_B64 (the
// bit-reversal permute is folded into the per-lane LDS dest address);
// the strain-row DMA overlaps the whole cum_h/searchsorted phase.
// ---------------------------------------------------------------------------

typedef float v2f __attribute__((ext_vector_type(2)));
typedef float v8f __attribute__((ext_vector_type(8)));

#define NTHREADS 256
#define FFTN     16384          // real samples per row
#define NC       8192           // complex FFT length (packed real FFT)
#define NFREQ    8193           // rfft bins
constexpr float PIF    = 3.14159265358979323846f;
constexpr float INV_SR = 1.0f / 2048.0f;   // 1/SAMPLE_RATE
constexpr float FOURDF = 0.5f;             // 4*DF, DF = 1/8

// LDS carve (bytes). No static __shared__ in this TU -> dynamic LDS base = 0,
// so these offsets are also the raw LDS byte addresses used by the async DMA.
constexpr unsigned OFF_BUF  = 0;                         // float2[8192]  = 65536
constexpr unsigned OFF_H    = OFF_BUF + 8192 * 8;        // float2[8200]  = 65600
constexpr unsigned OFF_CUM  = OFF_H + 8200 * 8;          // float[8256]   = 33024
constexpr unsigned OFF_SA   = OFF_CUM + 8256 * 4;        // float[256]
constexpr unsigned OFF_SB   = OFF_SA + 256 * 4;          // float[256]
constexpr unsigned OFF_CNT  = OFF_SB + 256 * 4;          // int[32]
constexpr unsigned OFF_EDG  = OFF_CNT + 32 * 4;          // int[32]
constexpr unsigned OFF_SHF  = OFF_EDG + 32 * 4;          // float[16]
constexpr unsigned SMEM_BYTES = OFF_SHF + 16 * 4;        // ~166.6 KB (fits 320KB WGP LDS)

__device__ __forceinline__ v8f wmma_f32(v2f a, v2f b, v8f c) {
    // D = A(16x4,f32) * B(4x16,f32) + C(16x16,f32); full fp32 matrix pipe.
    // F32 WMMA allows no A/B sign modifiers (NEG = {CNeg,0,0}) -> both false.
    return __builtin_amdgcn_wmma_f32_16x16x4_f32(
        /*neg_a=*/false, a, /*neg_b=*/false, b, /*c_mod=*/(short)0, c,
        /*reuse_a=*/false, /*reuse_b=*/false);
}

// Async DMA one row of 16384 reals into LDS as 8192 packed complex with the
// digit-reversed layout folded into the per-lane LDS destination address:
//   complex sample n = n1+16m (reals 2n and 2n+1 of the row) lands in bufc slot n1*512 + brev9(m)
// Tracked on ASYNCcnt; caller must wait_async_all() + barrier before use.
__device__ __forceinline__ void issue_row_async(const float* __restrict__ src, int tid) {
    for (int n = tid; n < NC; n += NTHREADS) {      // 32 issues/wave (< 63 cap)
        int n1 = n & 15;
        int m  = n >> 4;
        unsigned br = __brev((unsigned)m) >> 23;    // 9-bit bit reversal
        unsigned lds = OFF_BUF + (((((unsigned)n1) << 9) | br) << 3);
        unsigned long long ga = (unsigned long long)(const void*)(src + 2 * n);
        asm volatile("global_load_async_to_lds_b64 %0, %1, off"
                     :: "v"(lds), "v"(ga)
                     : "memory");
    }
}

__device__ __forceinline__ void wait_async_all() {
    asm volatile("s_wait_asynccnt 0" ::: "memory");
}

// 8192-pt complex FFT in LDS: 9 radix-2 stages inside 16 blocks of 512,
// then radix-16 recombine Z[512*k1+k2] = sum_n1 W8192^{n1 k2} W16^{n1 k1} Y[n1][k2]
// done as complex 16x16 DFT GEMM with V_WMMA_F32_16X16X4_F32.
__device__ __forceinline__ void fft8192(float2* bufc, int tid) {
    for (int s = 1; s <= 9; ++s) {
        int half = 1 << (s - 1);
        for (int b = tid; b < 4096; b += NTHREADS) {
            int blk = b >> 8;
            int idx = b & 255;
            int g = idx >> (s - 1);
            int p = idx & (half - 1);
            int i0 = (blk << 9) + g * (half << 1) + p;
            int i1 = i0 + half;
            float ang = -PIF * (float)p / (float)half;
            float sn, cs;
            __sincosf(ang, &sn, &cs);
            float2 a = bufc[i0], q = bufc[i1];
            float tr = q.x * cs - q.y * sn;
            float ti = q.x * sn + q.y * cs;
            bufc[i1] = make_float2(a.x - tr, a.y - ti);
            bufc[i0] = make_float2(a.x + tr, a.y + ti);
        }
        __syncthreads();
    }

    // ---- radix-16 recombine via fp32 WMMA (EXEC uniform here) ----
    const int lane = tid & 31;
    const int wave = tid >> 5;
    const int colL = lane & 15;
    const int koff = (lane < 16) ? 0 : 2;     // A/B K-half per ISA 32-bit layout
    const int mrow = lane & 15;               // A-matrix row (both lane halves)

    for (int tt = wave; tt < 32; tt += 8) {   // 32 column tiles of 16, 4 per wave
        const int col = (tt << 4) + colL;     // k2
        v2f are[4], aim[4], ain[4], bre[4], bim[4];
#pragma unroll
        for (int kb = 0; kb < 4; ++kb) {
#pragma unroll
            for (int u = 0; u < 2; ++u) {
                int n1 = (kb << 2) + koff + u;
                float2 y = bufc[(n1 << 9) + col];
                // B = Y twiddled by W8192^{n1*k2}
                float angb = -(2.0f * PIF / 8192.0f) * (float)(n1 * col);
                float sb_, cb_;
                __sincosf(angb, &sb_, &cb_);
                bre[kb][u] = y.x * cb_ - y.y * sb_;
                bim[kb][u] = y.x * sb_ + y.y * cb_;
                // A = DFT16 matrix: W16^{k1*n1}
                float anga = -(2.0f * PIF / 16.0f) * (float)((mrow * n1) & 15);
                float sa_, ca_;
                __sincosf(anga, &sa_, &ca_);
                are[kb][u] = ca_;
                aim[kb][u] = sa_;
                ain[kb][u] = -sa_;            // pre-negated Dim (no neg-A on f32 WMMA)
            }
        }
        v8f cre = {0.f, 0.f, 0.f, 0.f, 0.f, 0.f, 0.f, 0.f};
        v8f cim = {0.f, 0.f, 0.f, 0.f, 0.f, 0.f, 0.f, 0.f};
#pragma unroll
        for (int kb = 0; kb < 4; ++kb) {
            cre = wmma_f32(are[kb], bre[kb], cre);   // + Dre*Yre
            cre = wmma_f32(ain[kb], bim[kb], cre);   // - Dim*Yim
            cim = wmma_f32(are[kb], bim[kb], cim);   // + Dre*Yim
            cim = wmma_f32(aim[kb], bre[kb], cim);   // + Dim*Yre
        }
        const int rbase = (lane < 16) ? 0 : 8;       // C/D row layout
#pragma unroll
        for (int i = 0; i < 8; ++i) {
            int k1 = rbase + i;
            bufc[(k1 << 9) + col] = make_float2(cre[i], cim[i]);
        }
    }
    __syncthreads();
}

// rFFT untangle: X[k] for 0<=k<=8192 from packed FFT Z (unscaled)
__device__ __forceinline__ float2 untangle_at(const float2* bufc, int k) {
    float2 Zk = bufc[k & (NC - 1)];
    float2 Zm = bufc[(NC - k) & (NC - 1)];
    float ar = 0.5f * (Zk.x + Zm.x);
    float ai = 0.5f * (Zk.y - Zm.y);
    float br = 0.5f * (Zk.x - Zm.x);
    float bi = 0.5f * (Zk.y + Zm.y);
    float ang = -PIF * (float)k / (float)NC;
    float sn, cs;
    __sincosf(ang, &sn, &cs);
    return make_float2(ar + sn * br + cs * bi, ai + sn * bi - cs * br);
}

// Inclusive scan of cum[0..8192] (256 threads, 32-wide chunks + tail element)
__device__ __forceinline__ void block_scan(float* cum, float* sa, float* sb, int tid) {
    const int base = tid << 5;
    float s = 0.f;
    for (int i = 0; i < 32; ++i) s += cum[base + i];
    if (tid == NTHREADS - 1) s += cum[NC];
    sa[tid] = s;
    __syncthreads();
    float* src = sa;
    float* dst = sb;
    for (int ofs = 1; ofs < NTHREADS; ofs <<= 1) {
        float x = src[tid];
        if (tid >= ofs) x += src[tid - ofs];
        dst[tid] = x;
        __syncthreads();
        float* t = src; src = dst; dst = t;
    }
    float run = (tid == 0) ? 0.f : src[tid - 1];
    for (int i = 0; i < 32; ++i) {
        run += cum[base + i];
        cum[base + i] = run;
    }
    if (tid == NTHREADS - 1) {
        run += cum[NC];
        cum[NC] = run;
    }
    __syncthreads();
}

__global__ __launch_bounds__(NTHREADS, 1)
void ChiSq_773094113289_kernel(const float* __restrict__ tpl,
                               const float* __restrict__ strn,
                               float* __restrict__ out, int nrows) {
    extern __shared__ __align__(16) char smem[];
    float2* bufc = (float2*)(smem + OFF_BUF);
    float2* Hc   = (float2*)(smem + OFF_H);
    float*  cum  = (float*)(smem + OFF_CUM);
    float*  sa   = (float*)(smem + OFF_SA);
    float*  sb   = (float*)(smem + OFF_SB);
    int*    cnts = (int*)(smem + OFF_CNT);
    int*    edg  = (int*)(smem + OFF_EDG);
    float*  shf  = (float*)(smem + OFF_SHF);

    const int row = blockIdx.x;
    if (row >= nrows) return;
    const int tid = threadIdx.x;
    const float* trow = tpl + (size_t)row * FFTN;
    const float* srow = strn + (size_t)row * FFTN;

    // warm L2 with the strain row; its async DMA is issued much later
    __builtin_prefetch((const char*)srow + tid * 256, 0, 1);

    // ---- template FFT -> H, cum_h, edges ----
    issue_row_async(trow, tid);       // DMA + bit-reverse scatter into bufc
    wait_async_all();
    __syncthreads();
    fft8192(bufc, tid);
    for (int k = tid; k <= NC / 2; k += NTHREADS) {
        float2 Xk = untangle_at(bufc, k);
        Hc[k] = make_float2(Xk.x * INV_SR, Xk.y * INV_SR);
        int m = NC - k;
        float2 Xm = untangle_at(bufc, m);
        Hc[m] = make_float2(Xm.x * INV_SR, Xm.y * INV_SR);
    }
    __syncthreads();                  // bufc dead from here until strain FFT

    // overlap: strain-row DMA proceeds underneath the whole scan/edges phase
    issue_row_async(srow, tid);

    for (int k = tid; k <= NC; k += NTHREADS) {
        float2 h = Hc[k];
        cum[k] = FOURDF * (h.x * h.x + h.y * h.y);
    }
    __syncthreads();
    block_scan(cum, sa, sb, tid);
    if (tid == 0) shf[0] = cum[NC];
    __syncthreads();
    const float total_h = shf[0];

    if (tid < 17) cnts[tid] = 0;
    __syncthreads();
    {
        float binv[17];
#pragma unroll
        for (int j = 0; j < 17; ++j) binv[j] = total_h * (float)j * (1.0f / 16.0f);
        int loc[17];
#pragma unroll
        for (int j = 0; j < 17; ++j) loc[j] = 0;
        for (int k = tid; k <= NC; k += NTHREADS) {
            float v = cum[k];
#pragma unroll
            for (int j = 0; j < 17; ++j) loc[j] += (v <= binv[j]) ? 1 : 0;
        }
#pragma unroll
        for (int j = 0; j < 17; ++j) atomicAdd(&cnts[j], loc[j]);
    }
    __syncthreads();
    if (tid < 17) edg[tid] = (cnts[tid] > NC) ? NC : cnts[tid];   // clip to 8192
    __syncthreads();

    // ---- strain FFT -> cum_q, chi-square ----
    wait_async_all();                 // strain row now resident in bufc
    __syncthreads();
    fft8192(bufc, tid);
    const float isq = rsqrtf(total_h);
    for (int k = tid; k <= NC / 2; k += NTHREADS) {
        float2 Sk = untangle_at(bufc, k);
        float2 hk = Hc[k];
        cum[k] = FOURDF * (hk.x * (Sk.x * INV_SR) + hk.y * (Sk.y * INV_SR)) * isq;
        int m = NC - k;
        float2 Sm = untangle_at(bufc, m);
        float2 hm = Hc[m];
        cum[m] = FOURDF * (hm.x * (Sm.x * INV_SR) + hm.y * (Sm.y * INV_SR)) * isq;
    }
    __syncthreads();
    block_scan(cum, sa, sb, tid);

    if (tid == 0) {
        float total_snr = cum[NC];
        float mean = total_snr * (1.0f / 16.0f);
        float chisq = 0.f;
#pragma unroll
        for (int i = 0; i < 16; ++i) {
            float d = (cum[edg[i + 1]] - cum[edg[i]]) - mean;
            chisq += d * d;
        }
        out[row] = chisq * (16.0f / 15.0f);
    }
}

extern "C" void kernel_launch(void* const* d_in, const int* in_sizes, int n_in,
                              void* d_out, int out_size, void* d_ws, size_t ws_size,
                              hipStream_t stream) {
    const float* tpl  = (const float*)d_in[0];
    const float* strn = (const float*)d_in[1];
    float* out = (float*)d_out;
    int nrows = in_sizes[0] / FFTN;   // 512*2 = 1024 rows
    dim3 grid(nrows), block(NTHREADS);
    hipLaunchKernelGGL(ChiSq_773094113289_kernel, grid, block, SMEM_BYTES, stream,
                       tpl, strn, out, nrows);
}